// KalmanPredictorMambaAdjustment_14164802142682
// MI455X (gfx1250) — compile-verified
//
#include <hip/hip_runtime.h>
#include <hip/hip_bf16.h>
#include <math.h>

// ---------------------------------------------------------------------------
// SS2D (VMamba selective scan) x2 + combine, for MI455X (gfx1250, wave32).
// GEMMs use V_WMMA_F32_16X16X4_F32 (fp32 in/out, matrix pipe).
// The L=4096 recurrence is a 3-pass chunked parallel scan (16 chunks x 256).
// ---------------------------------------------------------------------------

#define D_MODEL 64
#define D_INNER 128
#define D_STATE 16
#define DT_RANK 4
#define KDIR    4
#define HH      64
#define WW      64
#define LL      4096
#define CPROJ   36            // DT_RANK + 2*D_STATE
#define NCHUNK  16
#define CHUNKL  256

typedef __attribute__((ext_vector_type(2))) float v2f;
typedef __attribute__((ext_vector_type(8))) float v8f;

__device__ __forceinline__ v8f wmma_f32_k4(v2f a, v2f b, v8f c) {
  // 8 args: (neg_a, A, neg_b, B, c_mod, C, reuse_a, reuse_b)
  return __builtin_amdgcn_wmma_f32_16x16x4_f32(false, a, false, b, (short)0, c,
                                               false, false);
}

__device__ __forceinline__ float silu_f(float v) {
  return v / (1.0f + __expf(-v));
}

// scan-position -> pixel index for direction k (H = W = 64)
__device__ __forceinline__ int scan_pixel(int k, int l) {
  int ll = (k & 2) ? (LL - 1 - l) : l;
  if (k & 1) return ((ll & 63) << 6) | (ll >> 6);  // (w,h)-major -> pixel
  return ll;
}

// ---------------------------------------------------------------------------
// K1: xz = Xpix[4096x64] * W_in[64x256] + b_in ; split -> xq (ch-major), z
// one wave per 16x16 tile; K=64 -> 16 WMMA f32 k4 steps
// ---------------------------------------------------------------------------
__global__ void k_inproj(const float* __restrict__ x,     // [64][L]
                         const float* __restrict__ W_in,  // [64][256]
                         const float* __restrict__ b_in,  // [256]
                         float* __restrict__ xq,          // [128][L]
                         float* __restrict__ z) {         // [L][128]
  const int lane = threadIdx.x & 31;
  const int half = lane >> 4;
  const int lid  = lane & 15;
  const int row0 = blockIdx.x * 16;   // pixel tile
  const int col0 = blockIdx.y * 16;   // output-channel tile
  v8f acc = {};
#pragma unroll
  for (int kb = 0; kb < D_MODEL; kb += 4) {
    const int ka = kb + 2 * half;
    v2f a, b;
    a.x = x[(ka + 0) * LL + row0 + lid];
    a.y = x[(ka + 1) * LL + row0 + lid];
    b.x = W_in[(ka + 0) * 256 + col0 + lid];
    b.y = W_in[(ka + 1) * 256 + col0 + lid];
    acc = wmma_f32_k4(a, b, acc);
  }
  const float bias = b_in[col0 + lid];
  const int col = col0 + lid;
#pragma unroll
  for (int r = 0; r < 8; ++r) {
    const int p = row0 + r + 8 * half;
    const float v = acc[r] + bias;
    if (col < D_INNER) xq[col * LL + p] = v;
    else               z[p * D_INNER + (col - D_INNER)] = v;
  }
}

// ---------------------------------------------------------------------------
// K2: depthwise 3x3 SAME conv + SiLU : xc[d][p]
// ---------------------------------------------------------------------------
__global__ void k_dwconv(const float* __restrict__ xq,
                         const float* __restrict__ conv_w,  // [128][1][3][3]
                         const float* __restrict__ conv_b,  // [128]
                         float* __restrict__ xc) {
  const int idx = blockIdx.x * blockDim.x + threadIdx.x;  // d*L + p
  const int d = idx >> 12;
  const int p = idx & (LL - 1);
  const int h = p >> 6, w = p & 63;
  float s = conv_b[d];
#pragma unroll
  for (int i = 0; i < 3; ++i) {
#pragma unroll
    for (int j = 0; j < 3; ++j) {
      const int hh = h + i - 1, ww = w + j - 1;
      if (hh >= 0 && hh < HH && ww >= 0 && ww < WW)
        s += conv_w[d * 9 + i * 3 + j] * xq[d * LL + hh * WW + ww];
    }
  }
  xc[idx] = silu_f(s);
}

// ---------------------------------------------------------------------------
// K3: pdbl[k][c][p] = sum_d xproj_w[k][c][d] * xc[d][p]   (c padded 36->48)
// grid: (3 Mtiles, 256 Ntiles, 4 k) ; K=128 -> 32 WMMA steps
// ---------------------------------------------------------------------------
__global__ void k_xproj(const float* __restrict__ xc,
                        const float* __restrict__ xproj_w,  // [4][36][128]
                        float* __restrict__ pdbl) {         // [4][36][L]
  const int lane = threadIdx.x & 31;
  const int half = lane >> 4;
  const int lid  = lane & 15;
  const int row0 = blockIdx.x * 16;
  const int col0 = blockIdx.y * 16;
  const int k    = blockIdx.z;
  const float* Wk = xproj_w + k * CPROJ * D_INNER;
  const int m = row0 + lid;
  v8f acc = {};
#pragma unroll 4
  for (int kb = 0; kb < D_INNER; kb += 4) {
    const int ka = kb + 2 * half;
    v2f a, b;
    a.x = (m < CPROJ) ? Wk[m * D_INNER + ka + 0] : 0.0f;
    a.y = (m < CPROJ) ? Wk[m * D_INNER + ka + 1] : 0.0f;
    b.x = xc[(ka + 0) * LL + col0 + lid];
    b.y = xc[(ka + 1) * LL + col0 + lid];
    acc = wmma_f32_k4(a, b, acc);
  }
#pragma unroll
  for (int r = 0; r < 8; ++r) {
    const int c = row0 + r + 8 * half;
    if (c < CPROJ) pdbl[(k * CPROJ + c) * LL + col0 + lid] = acc[r];
  }
}

// ---------------------------------------------------------------------------
// K4: pdelta[k][d][p] = softplus(dt_w[k] (128x4) @ dts (4xL) + dt_b[k][d])
// single WMMA f32 16x16x4 per tile (K == DT_RANK == 4)
// ---------------------------------------------------------------------------
__global__ void k_delta(const float* __restrict__ pdbl,
                        const float* __restrict__ dt_w,   // [4][128][4]
                        const float* __restrict__ dt_b,   // [4][128]
                        float* __restrict__ pdelta) {     // [4][128][L]
  const int lane = threadIdx.x & 31;
  const int half = lane >> 4;
  const int lid  = lane & 15;
  const int row0 = blockIdx.x * 16;   // d tile
  const int col0 = blockIdx.y * 16;   // p tile
  const int k    = blockIdx.z;
  const int ka = 2 * half;            // 0 or 2
  const int m  = row0 + lid;
  v2f a, b;
  a.x = dt_w[(k * D_INNER + m) * DT_RANK + ka + 0];
  a.y = dt_w[(k * D_INNER + m) * DT_RANK + ka + 1];
  b.x = pdbl[(k * CPROJ + ka + 0) * LL + col0 + lid];
  b.y = pdbl[(k * CPROJ + ka + 1) * LL + col0 + lid];
  v8f acc = {};
  acc = wmma_f32_k4(a, b, acc);
#pragma unroll
  for (int r = 0; r < 8; ++r) {
    const int d = row0 + r + 8 * half;
    const float v = acc[r] + dt_b[k * D_INNER + d];
    const float sp = (v > 20.0f) ? v : log1pf(__expf(v));
    pdelta[(k * D_INNER + d) * LL + col0 + lid] = sp;
  }
}

// ---------------------------------------------------------------------------
// Scan pass A: per (k,d,chunk) x 16 states, compute chunk carry (prodA, aggB)
// thread: n = tid&15, unit = (k*128+d)*16 + chunk
// ---------------------------------------------------------------------------
__global__ void k_scanA(const float* __restrict__ pdelta,
                        const float* __restrict__ pdbl,
                        const float* __restrict__ xc,
                        const float* __restrict__ A_logs,  // [512][16]
                        float* __restrict__ carryA,
                        float* __restrict__ carryB) {
  const int t = blockIdx.x * blockDim.x + threadIdx.x;
  const int n = t & 15;
  const int unit = t >> 4;
  const int chunk = unit & (NCHUNK - 1);
  const int kd = unit >> 4;             // k*128 + d
  const int d = kd & (D_INNER - 1), k = kd >> 7;
  const float An = -__expf(A_logs[kd * D_STATE + n]);
  const float* dp = pdelta + kd * LL;
  const float* Bp = pdbl + (k * CPROJ + DT_RANK + n) * LL;
  const float* up = xc + d * LL;
  float Aprod = 1.0f, Bagg = 0.0f;
  const int l0 = chunk * CHUNKL;
  for (int i = 0; i < CHUNKL; ++i) {
    const int p = scan_pixel(k, l0 + i);
    if ((i & 31) == 0) {
      const int pf = scan_pixel(k, l0 + ((i + 32) & (CHUNKL - 1)));
      __builtin_prefetch(dp + pf, 0, 1);   // global_prefetch_b8
      __builtin_prefetch(Bp + pf, 0, 1);
    }
    const float delta = dp[p];
    const float a  = __expf(delta * An);
    const float bu = delta * Bp[p] * up[p];
    Aprod *= a;
    Bagg = a * Bagg + bu;
  }
  carryA[unit * D_STATE + n] = Aprod;
  carryB[unit * D_STATE + n] = Bagg;
}

// ---------------------------------------------------------------------------
// Scan pass B: sequential combine over the 16 chunks -> chunk-entry states
// ---------------------------------------------------------------------------
__global__ void k_scanB(const float* __restrict__ carryA,
                        const float* __restrict__ carryB,
                        float* __restrict__ hin) {
  const int t = blockIdx.x * blockDim.x + threadIdx.x;  // kd*16+n, 8192 total
  if (t >= KDIR * D_INNER * D_STATE) return;
  const int n = t & 15, kd = t >> 4;
  float h = 0.0f;
#pragma unroll
  for (int c = 0; c < NCHUNK; ++c) {
    const int unit = kd * NCHUNK + c;
    hin[unit * D_STATE + n] = h;
    h = carryA[unit * D_STATE + n] * h + carryB[unit * D_STATE + n];
  }
}

// ---------------------------------------------------------------------------
// Scan pass C: replay with correct entry state, emit y = sum_n h*C + Ds*u,
// accumulating the cross-scan merge directly in pixel space via f32 atomics.
// ---------------------------------------------------------------------------
__global__ void k_scanC(const float* __restrict__ pdelta,
                        const float* __restrict__ pdbl,
                        const float* __restrict__ xc,
                        const float* __restrict__ A_logs,
                        const float* __restrict__ Ds,      // [512]
                        const float* __restrict__ hin,
                        float* __restrict__ ymerge) {      // [128][L], zeroed
  const int t = blockIdx.x * blockDim.x + threadIdx.x;
  const int n = t & 15;
  const int unit = t >> 4;
  const int chunk = unit & (NCHUNK - 1);
  const int kd = unit >> 4;
  const int d = kd & (D_INNER - 1), k = kd >> 7;
  const float An = -__expf(A_logs[kd * D_STATE + n]);
  const float Dsv = Ds[kd];
  const float* dp = pdelta + kd * LL;
  const float* Bp = pdbl + (k * CPROJ + DT_RANK + n) * LL;
  const float* Cp = pdbl + (k * CPROJ + DT_RANK + D_STATE + n) * LL;
  const float* up = xc + d * LL;
  float h = hin[unit * D_STATE + n];
  const int l0 = chunk * CHUNKL;
  for (int i = 0; i < CHUNKL; ++i) {
    const int p = scan_pixel(k, l0 + i);
    if ((i & 31) == 0) {
      const int pf = scan_pixel(k, l0 + ((i + 32) & (CHUNKL - 1)));
      __builtin_prefetch(dp + pf, 0, 1);
      __builtin_prefetch(Cp + pf, 0, 1);
    }
    const float delta = dp[p];
    const float u = up[p];
    const float a = __expf(delta * An);
    h = a * h + delta * Bp[p] * u;
    float part = h * Cp[p];
    part += __shfl_xor(part, 1);
    part += __shfl_xor(part, 2);
    part += __shfl_xor(part, 4);
    part += __shfl_xor(part, 8);
    if (n == 0) atomicAdd(&ymerge[d * LL + p], part + Dsv * u);
  }
}

// ---------------------------------------------------------------------------
// K6: LayerNorm over d, SiLU gate with z, -> g[p][128] (row-major for GEMM A)
// ---------------------------------------------------------------------------
__global__ void k_lngate(const float* __restrict__ ymerge,  // [128][L]
                         const float* __restrict__ z,       // [L][128]
                         const float* __restrict__ ln_w,
                         const float* __restrict__ ln_b,
                         float* __restrict__ g) {           // [L][128]
  const int p = blockIdx.x * blockDim.x + threadIdx.x;      // 4096 pixels
  float s = 0.0f, ss = 0.0f;
  for (int d = 0; d < D_INNER; ++d) {
    const float v = ymerge[d * LL + p];
    s += v; ss += v * v;
  }
  const float mu = s * (1.0f / D_INNER);
  const float var = ss * (1.0f / D_INNER) - mu * mu;
  const float rs = rsqrtf(var + 1e-5f);
  for (int d = 0; d < D_INNER; ++d) {
    const float v = ymerge[d * LL + p];
    const float zz = z[p * D_INNER + d];
    g[p * D_INNER + d] = ((v - mu) * rs * ln_w[d] + ln_b[d]) * silu_f(zz);
  }
}

// ---------------------------------------------------------------------------
// K7: output projection: oproj[c][p] = g[4096x128] @ W_out[128x64] + b_out
// ---------------------------------------------------------------------------
__global__ void k_outproj(const float* __restrict__ g,
                          const float* __restrict__ W_out,  // [128][64]
                          const float* __restrict__ b_out,  // [64]
                          float* __restrict__ oproj) {      // [64][L]
  const int lane = threadIdx.x & 31;
  const int half = lane >> 4;
  const int lid  = lane & 15;
  const int row0 = blockIdx.x * 16;   // pixel tile
  const int col0 = blockIdx.y * 16;   // out-channel tile
  v8f acc = {};
#pragma unroll 4
  for (int kb = 0; kb < D_INNER; kb += 4) {
    const int ka = kb + 2 * half;
    v2f a, b;
    a.x = g[(row0 + lid) * D_INNER + ka + 0];
    a.y = g[(row0 + lid) * D_INNER + ka + 1];
    b.x = W_out[(ka + 0) * D_MODEL + col0 + lid];
    b.y = W_out[(ka + 1) * D_MODEL + col0 + lid];
    acc = wmma_f32_k4(a, b, acc);
  }
  const float bias = b_out[col0 + lid];
  const int c = col0 + lid;
#pragma unroll
  for (int r = 0; r < 8; ++r) {
    const int p = row0 + r + 8 * half;
    oproj[c * LL + p] = acc[r] + bias;
  }
}

// ---------------------------------------------------------------------------
// K8: out = x * exp(sigma) + bias   (x / exp(-sigma) == x * exp(sigma))
// ---------------------------------------------------------------------------
__global__ void k_combine(const float* __restrict__ x,
                          const float* __restrict__ sig,
                          const float* __restrict__ bia,
                          float* __restrict__ out) {
  const int i = blockIdx.x * blockDim.x + threadIdx.x;
  if (i < D_MODEL * LL) out[i] = x[i] * __expf(sig[i]) + bia[i];
}

// ---------------------------------------------------------------------------
// Workspace layout (floats); both param sets share the intermediates.
// ---------------------------------------------------------------------------
#define OFF_XQ      0u
#define OFF_Z       524288u
#define OFF_XC      1048576u
#define OFF_PDBL    1572864u     // 4*36*4096 = 589824
#define OFF_PDELTA  2162688u     // 4*128*4096 = 2097152
#define OFF_CARRYA  4259840u     // 4*128*16*16 = 131072
#define OFF_CARRYB  4390912u
#define OFF_HIN     4521984u
#define OFF_YMERGE  4653056u     // 128*4096
#define OFF_G       5177344u     // 4096*128
#define OFF_OPROJ0  5701632u     // 64*4096
#define OFF_OPROJ1  5963776u     // 64*4096   (end = 6225920 floats ~ 24.9 MB)

extern "C" void kernel_launch(void* const* d_in, const int* in_sizes, int n_in,
                              void* d_out, int out_size, void* d_ws,
                              size_t ws_size, hipStream_t stream) {
  (void)in_sizes; (void)n_in; (void)out_size; (void)ws_size;
  const float* x = (const float*)d_in[0];
  float* ws = (float*)d_ws;
  float* xq     = ws + OFF_XQ;
  float* z      = ws + OFF_Z;
  float* xc     = ws + OFF_XC;
  float* pdbl   = ws + OFF_PDBL;
  float* pdelta = ws + OFF_PDELTA;
  float* carryA = ws + OFF_CARRYA;
  float* carryB = ws + OFF_CARRYB;
  float* hin    = ws + OFF_HIN;
  float* ymerge = ws + OFF_YMERGE;
  float* g      = ws + OFF_G;
  float* oproj[2] = {ws + OFF_OPROJ0, ws + OFF_OPROJ1};

  for (int set = 0; set < 2; ++set) {
    const int base = 1 + set * 13;
    const float* W_in    = (const float*)d_in[base + 0];
    const float* b_in    = (const float*)d_in[base + 1];
    const float* conv_w  = (const float*)d_in[base + 2];
    const float* conv_b  = (const float*)d_in[base + 3];
    const float* xproj_w = (const float*)d_in[base + 4];
    const float* dt_w    = (const float*)d_in[base + 5];
    const float* dt_b    = (const float*)d_in[base + 6];
    const float* A_logs  = (const float*)d_in[base + 7];
    const float* Ds      = (const float*)d_in[base + 8];
    const float* ln_w    = (const float*)d_in[base + 9];
    const float* ln_b    = (const float*)d_in[base + 10];
    const float* W_out   = (const float*)d_in[base + 11];
    const float* b_out   = (const float*)d_in[base + 12];

    hipMemsetAsync(ymerge, 0, (size_t)D_INNER * LL * sizeof(float), stream);
    k_inproj <<<dim3(LL / 16, 256 / 16), 32, 0, stream>>>(x, W_in, b_in, xq, z);
    k_dwconv <<<(D_INNER * LL) / 256, 256, 0, stream>>>(xq, conv_w, conv_b, xc);
    k_xproj  <<<dim3(3, LL / 16, KDIR), 32, 0, stream>>>(xc, xproj_w, pdbl);
    k_delta  <<<dim3(D_INNER / 16, LL / 16, KDIR), 32, 0, stream>>>(pdbl, dt_w,
                                                                    dt_b, pdelta);
    // 8192 units * 16 state-threads = 131072 threads
    k_scanA  <<<1024, 128, 0, stream>>>(pdelta, pdbl, xc, A_logs, carryA, carryB);
    k_scanB  <<<64, 128, 0, stream>>>(carryA, carryB, hin);
    k_scanC  <<<1024, 128, 0, stream>>>(pdelta, pdbl, xc, A_logs, Ds, hin, ymerge);
    k_lngate <<<LL / 256, 256, 0, stream>>>(ymerge, z, ln_w, ln_b, g);
    k_outproj<<<dim3(LL / 16, D_MODEL / 16), 32, 0, stream>>>(g, W_out, b_out,
                                                              oproj[set]);
  }
  k_combine<<<(D_MODEL * LL + 255) / 256, 256, 0, stream>>>(x, oproj[0],
                                                            oproj[1],
                                                            (float*)d_out);
}